// RotaryPositionalEmbedding_12859132084556
// MI455X (gfx1250) — compile-verified
//
#include <hip/hip_runtime.h>
#include <math.h>

// RoPE: x (4, 16, 4096, 64) fp32, theta = 10000, d_k = 64.
// Reference ignores token_positions (rebuilds arange), so we do too.
//
// Memory-bound op: 128 MiB total traffic -> ~5.7 us floor at 23.3 TB/s.
// Strategy:
//   1. Tiny kernel builds a (cos,sin) table for the 4096 x 32 unique angles
//      (1 MiB) using accurate sincosf (angles up to 4096 rad need full
//      range reduction; hardware v_sin_f32 alone is out of range).
//   2. Main kernel: one float4 of x per thread = 2 rotation pairs.
//      Non-temporal b128 load/store for the streamed x/out so the hot
//      1 MiB table stays resident in the 192 MB L2 (each table entry is
//      reused 64x across batch*head).
//
// rope_apply placed first in the file so its device asm shows up in the
// feedback snippet (verifying TH_NT hints on the b128 load/store).

typedef float v4f __attribute__((ext_vector_type(4)));
typedef float v2f __attribute__((ext_vector_type(2)));

#define SEQ    4096
#define NPAIR  32          // D_K / 2
#define TBL_ELEMS (SEQ * NPAIR)            // float2 entries
#define TBL_BYTES (TBL_ELEMS * 2 * 4)      // 1 MiB

// log2(10000)
#define LOG2_THETA 13.287712379549449f

__global__ void rope_apply(const v4f* __restrict__ x,
                           const v4f* __restrict__ tab,   // [SEQ][16] v4f
                           v4f* __restrict__ out,
                           int total4) {
    int i = blockIdx.x * blockDim.x + threadIdx.x;
    if (i >= total4) return;
    int j = i & 15;                 // which float4 within the 64-wide head dim
    int s = (i >> 4) & (SEQ - 1);   // sequence position

    // Streamed data: non-temporal so it doesn't evict the table from L2.
    v4f xv = __builtin_nontemporal_load(&x[i]);
    // Table: regular temporal load; 1 MiB working set stays L2-resident.
    v4f t = tab[s * 16 + j];        // (cos0, sin0, cos1, sin1)

    v4f o;
    o.x = t.x * xv.x - t.y * xv.y;  // cos*even - sin*odd
    o.y = t.y * xv.x + t.x * xv.y;  // sin*even + cos*odd
    o.z = t.z * xv.z - t.w * xv.w;
    o.w = t.w * xv.z + t.z * xv.w;

    __builtin_nontemporal_store(o, &out[i]);
}

__global__ void rope_build_table(v2f* __restrict__ tab) {
    int idx = blockIdx.x * blockDim.x + threadIdx.x;
    if (idx >= TBL_ELEMS) return;
    int k = idx & (NPAIR - 1);     // pair index 0..31
    int s = idx >> 5;              // position 0..4095
    // inv_freq = theta^(-k/32) = exp2(-k * log2(theta) / 32)
    float inv = exp2f(-(float)k * (LOG2_THETA / 32.0f));
    float ang = (float)s * inv;
    float sv, cv;
    sincosf(ang, &sv, &cv);        // accurate: full range reduction
    v2f t; t.x = cv; t.y = sv;
    tab[idx] = t;
}

// Fallback if workspace is too small for the table: fuse sincos inline.
__global__ void rope_fused(const v4f* __restrict__ x,
                           v4f* __restrict__ out,
                           int total4) {
    int i = blockIdx.x * blockDim.x + threadIdx.x;
    if (i >= total4) return;
    int j = i & 15;
    int s = (i >> 4) & (SEQ - 1);
    int k0 = 2 * j;
    float inv0 = exp2f(-(float)k0       * (LOG2_THETA / 32.0f));
    float inv1 = exp2f(-(float)(k0 + 1) * (LOG2_THETA / 32.0f));
    float s0, c0, s1, c1;
    sincosf((float)s * inv0, &s0, &c0);
    sincosf((float)s * inv1, &s1, &c1);

    v4f xv = __builtin_nontemporal_load(&x[i]);
    v4f o;
    o.x = c0 * xv.x - s0 * xv.y;
    o.y = s0 * xv.x + c0 * xv.y;
    o.z = c1 * xv.z - s1 * xv.w;
    o.w = s1 * xv.z + c1 * xv.w;
    __builtin_nontemporal_store(o, &out[i]);
}

extern "C" void kernel_launch(void* const* d_in, const int* in_sizes, int n_in,
                              void* d_out, int out_size, void* d_ws, size_t ws_size,
                              hipStream_t stream) {
    const v4f* x = (const v4f*)d_in[0];
    v4f* out = (v4f*)d_out;
    const int total4 = out_size / 4;           // 4,194,304 float4 lanes
    const int block = 256;

    if (ws_size >= (size_t)TBL_BYTES && d_ws != nullptr) {
        v2f* tab = (v2f*)d_ws;
        int tgrid = (TBL_ELEMS + block - 1) / block;
        rope_build_table<<<tgrid, block, 0, stream>>>(tab);
        int grid = (total4 + block - 1) / block;
        rope_apply<<<grid, block, 0, stream>>>(x, (const v4f*)tab, out, total4);
    } else {
        int grid = (total4 + block - 1) / block;
        rope_fused<<<grid, block, 0, stream>>>(x, out, total4);
    }
}